// PoolLogSumExp_71665824301167
// MI455X (gfx1250) — compile-verified
//
#include <hip/hip_runtime.h>
#include <hip/hip_bf16.h>
#include <stdint.h>

#define TILE_ROWS 16                       // rows per tile (16KB @ D=256)
#define NBUF 4                             // LDS buffers (64KB total)
#define PF 3                               // prefetch depth (tiles in flight)
#define OPT ((TILE_ROWS * 256) / (4 * 256))  // async b128 ops per thread per tile
static_assert(OPT == 4, "wait immediates below assume 4 async ops/thread/tile");
static_assert(NBUF > PF, "buffer reuse hazard: need NBUF > PF");
#define LSE_EPS 1e-12f

// Online logsumexp accumulator update (branchless; exp(-inf)=0 handles init).
static __device__ __forceinline__ void lse_update(float x, float& m, float& s) {
    float nm = fmaxf(m, x);
    s = s * __expf(m - nm) + __expf(x - nm);
    m = nm;
}

// ---------------------------------------------------------------------------
// d_feat == 256 kernel. One block (256 threads = 8 wave32) per segment.
// 16-row tiles streamed into a 4-deep LDS ring via global_load_async_to_lds,
// 3 tiles in flight per wave (partial s_wait_asynccnt), online LSE per column.
// ---------------------------------------------------------------------------
__global__ __launch_bounds__(256) void seg_lse_async256(
    const float* __restrict__ feats, const int* __restrict__ batch,
    float* __restrict__ out, int n_nodes)
{
    constexpr int D = 256;
    __shared__ float smem[NBUF][TILE_ROWS * D];   // exactly 64KB

    const int g = blockIdx.x;
    const int t = threadIdx.x;

    // Segment bounds: redundant per-thread binary search (uniform, cheap).
    int lo = 0, hi = n_nodes;
    while (lo < hi) { int mid = (lo + hi) >> 1; if (batch[mid] < g) lo = mid + 1; else hi = mid; }
    const int start = lo;
    hi = n_nodes;
    while (lo < hi) { int mid = (lo + hi) >> 1; if (batch[mid] < g + 1) lo = mid + 1; else hi = mid; }
    const int end   = lo;
    const int nrows = end - start;

    float m0 = -__builtin_inff(), s0 = 0.0f;   // even rows
    float m1 = -__builtin_inff(), s1 = 0.0f;   // odd rows

    if (nrows > 0) {
        const int ntiles = (nrows + TILE_ROWS - 1) / TILE_ROWS;

        // Issue async copies of tile `tile` into ring buffer p (OPT ops/thread).
        auto issue_tile = [&](int tile, int p) {
            uint32_t lds_base = (uint32_t)(uintptr_t)(&smem[p][0]);
            const int row0 = start + tile * TILE_ROWS;
#pragma unroll
            for (int k = 0; k < OPT; ++k) {
                int c   = t + k * 256;        // 16B chunk id within tile
                int fl  = c << 2;             // flat float index within tile
                int r   = fl >> 8;            // fl / 256
                int col = fl & (D - 1);
                int row = row0 + r;
                row = (row < end - 1) ? row : (end - 1);   // clamp tail rows
                const float* gp = feats + (size_t)row * D + col;
                uint32_t lds = lds_base + (uint32_t)(fl << 2);
                asm volatile("global_load_async_to_lds_b128 %0, %1, off"
                             :: "v"(lds), "v"(gp)
                             : "memory");
            }
        };

        // Prologue: launch up to PF tiles.
        const int pre = (ntiles < PF) ? ntiles : PF;
        for (int i = 0; i < pre; ++i) issue_tile(i, i);

        for (int k = 0; k < ntiles; ++k) {
            // Tiles issued beyond tile k (per wave): rem in {0,1,2}.
            int rem = ntiles - k - 1;
            if (rem > PF - 1) rem = PF - 1;
            if (rem == 0)      asm volatile("s_wait_asynccnt 0" ::: "memory");
            else if (rem == 1) asm volatile("s_wait_asynccnt 4" ::: "memory");
            else               asm volatile("s_wait_asynccnt 8" ::: "memory");
            __syncthreads();   // all waves: tile k landed, old buffer fully read

            if (k + PF < ntiles) issue_tile(k + PF, (k + PF) % NBUF);

            const float* buf = &smem[k % NBUF][0];
            const int rows_this = min(TILE_ROWS, nrows - k * TILE_ROWS);
            if (rows_this == TILE_ROWS) {
#pragma unroll
                for (int r = 0; r < TILE_ROWS; r += 2) {
                    lse_update(buf[r * D + t],       m0, s0);
                    lse_update(buf[(r + 1) * D + t], m1, s1);
                }
            } else {
                for (int r = 0; r < rows_this; r += 2) {
                    lse_update(buf[r * D + t], m0, s0);
                    if (r + 1 < rows_this) lse_update(buf[(r + 1) * D + t], m1, s1);
                }
            }

            asm volatile("s_wait_dscnt 0" ::: "memory");   // LDS reads done before
                                                           // next barrier permits reuse
        }
    }

    // Merge accumulators; empty segment -> log(EPS) + 0 (reference semantics).
    float m = fmaxf(m0, m1);
    float s;
    if (m == -__builtin_inff()) { m = 0.0f; s = 0.0f; }
    else s = s0 * __expf(m0 - m) + s1 * __expf(m1 - m);
    out[(size_t)g * D + t] = logf(fmaxf(s, LSE_EPS)) + m;
}

// ---------------------------------------------------------------------------
// Generic fallback for d_feat != 256 (plain coalesced global loads).
// ---------------------------------------------------------------------------
__global__ void seg_lse_generic(
    const float* __restrict__ feats, const int* __restrict__ batch,
    float* __restrict__ out, int n_nodes, int d_feat)
{
    const int g = blockIdx.x;
    const int col = blockIdx.y * blockDim.x + threadIdx.x;
    if (col >= d_feat) return;

    int lo = 0, hi = n_nodes;
    while (lo < hi) { int mid = (lo + hi) >> 1; if (batch[mid] < g) lo = mid + 1; else hi = mid; }
    const int start = lo;
    hi = n_nodes;
    while (lo < hi) { int mid = (lo + hi) >> 1; if (batch[mid] < g + 1) lo = mid + 1; else hi = mid; }
    const int end = lo;

    float m = -__builtin_inff(), s = 0.0f;
    for (int r = start; r < end; ++r)
        lse_update(feats[(size_t)r * d_feat + col], m, s);

    if (m == -__builtin_inff()) { m = 0.0f; s = 0.0f; }
    out[(size_t)g * d_feat + col] = logf(fmaxf(s, LSE_EPS)) + m;
}

extern "C" void kernel_launch(void* const* d_in, const int* in_sizes, int n_in,
                              void* d_out, int out_size, void* d_ws, size_t ws_size,
                              hipStream_t stream) {
    (void)n_in; (void)d_ws; (void)ws_size;
    const float* feats = (const float*)d_in[0];
    const int*   batch = (const int*)d_in[1];
    float*       out   = (float*)d_out;

    const int n_nodes = in_sizes[1];
    const int d_feat  = (n_nodes > 0) ? (in_sizes[0] / n_nodes) : 1;
    const int n_seg   = (d_feat > 0) ? (out_size / d_feat) : 0;
    if (n_seg <= 0) return;

    if (d_feat == 256) {
        seg_lse_async256<<<n_seg, 256, 0, stream>>>(feats, batch, out, n_nodes);
    } else {
        dim3 grid(n_seg, (d_feat + 255) / 256);
        seg_lse_generic<<<grid, 256, 0, stream>>>(feats, batch, out, n_nodes, d_feat);
    }
}